// MultiHeadAttention_52261162058330
// MI455X (gfx1250) — compile-verified
//
#include <hip/hip_runtime.h>

// ---------------------------------------------------------------------------
// out = (X X^T) X  ==  X (X^T X)   (matmul associativity; no softmax present)
// Per (b,h) slice: X is [2048, 64] fp32.
//   Phase 1: P[bh][slice] = partial X^T X over a 512-row j-slice  (64x64 fp32)
//   Phase 2: M = sum(P), out = X @ M
// All matrix math via v_wmma_f32_16x16x4_f32 (CDNA5 fp32 WMMA, wave32).
// ---------------------------------------------------------------------------

typedef __attribute__((ext_vector_type(2))) float v2f;
typedef __attribute__((ext_vector_type(8))) float v8f;

#define BH      32
#define TLEN    2048
#define DIM     64
#define SLICES  4
#define JS      (TLEN / SLICES)   // 512 rows per slice
#define LDSW    72                // LDS row stride (banks: +0/+16 and +8/+24 sets)

// ---------------------------------------------------------------------------
// Phase 1: one block per (bh, slice); 16 waves; wave w owns tile (et,dt) of
// the 64x64 X^T X partial. A-matrix = X^T (16x4 per step), B-matrix = X (4x16).
// 32-bit A/B VGPR layout (ISA 7.12.2): lanes 0-15 hold K=0,1 ; lanes 16-31 K=2,3.
// ---------------------------------------------------------------------------
__global__ __launch_bounds__(512)
void xtx_partial_kernel(const float* __restrict__ x, float* __restrict__ P)
{
    const int bh    = blockIdx.x >> 2;
    const int slice = blockIdx.x & 3;
    const int wave  = threadIdx.x >> 5;     // wave32
    const int lane  = threadIdx.x & 31;
    const int half  = lane >> 4;            // 0: K pair {0,1}, 1: K pair {2,3}
    const int l     = lane & 15;
    const int et    = wave >> 2;            // e-tile (rows of X^T X)
    const int dt    = wave & 3;             // d-tile (cols of X^T X)

    const float* __restrict__ X = x + (size_t)bh * TLEN * DIM;
    const int ecol = et * 16 + l;           // A: M index = column e of X
    const int dcol = dt * 16 + l;           // B: N index = column d of X

    v8f acc0 = {};                          // two accumulators -> break WMMA RAW chain
    v8f acc1 = {};

    const int j0 = slice * JS;
    #pragma unroll 4
    for (int j = j0; j < j0 + JS; j += 8) {
        // --- k-step 0: rows j .. j+3 ---
        {
            const int jA = j + 2 * half;
            v2f a, b;
            a.x = X[jA * DIM + ecol];
            a.y = X[(jA + 1) * DIM + ecol];
            b.x = X[jA * DIM + dcol];
            b.y = X[(jA + 1) * DIM + dcol];
            acc0 = __builtin_amdgcn_wmma_f32_16x16x4_f32(false, a, false, b,
                                                         (short)0, acc0, false, false);
        }
        // --- k-step 1: rows j+4 .. j+7 ---
        {
            const int jA = j + 4 + 2 * half;
            v2f a, b;
            a.x = X[jA * DIM + ecol];
            a.y = X[(jA + 1) * DIM + ecol];
            b.x = X[jA * DIM + dcol];
            b.y = X[(jA + 1) * DIM + dcol];
            acc1 = __builtin_amdgcn_wmma_f32_16x16x4_f32(false, a, false, b,
                                                         (short)0, acc1, false, false);
        }
    }
    const v8f acc = acc0 + acc1;

    // D layout (ISA 7.12.2): lane<16 -> VGPR r holds M=r ; lane>=16 -> M=8+r ; N = l
    float* __restrict__ Pout = P + (size_t)(bh * SLICES + slice) * DIM * DIM;
    #pragma unroll
    for (int r = 0; r < 8; ++r) {
        const int e = et * 16 + r + 8 * half;
        Pout[e * DIM + dcol] = acc[r];
    }
}

// ---------------------------------------------------------------------------
// Phase 2: block = 8 waves covering 8 j-tiles of one bh slice.
// Pre-sum the 4 partials of M=X^T X into LDS (stride-72 rows to keep the two
// lane-halves of each ds_load_b32 on disjoint bank sets), then each wave does
// out[16 rows] = X_tile @ M with 4 d-tile accumulators, K=64 in steps of 4.
// ---------------------------------------------------------------------------
__global__ __launch_bounds__(256)
void xm_kernel(const float* __restrict__ x, const float* __restrict__ P,
               float* __restrict__ out)
{
    __shared__ float Ms[DIM * LDSW];

    const int bh   = blockIdx.x >> 4;
    const int jtb  = (blockIdx.x & 15) * 8;
    const int wave = threadIdx.x >> 5;
    const int lane = threadIdx.x & 31;
    const int half = lane >> 4;
    const int l    = lane & 15;

    // Reduce the 4 slice-partials into LDS (coalesced along d).
    const float* __restrict__ Pb = P + (size_t)bh * SLICES * DIM * DIM;
    for (int i = threadIdx.x; i < DIM * DIM; i += 256) {
        const float s = Pb[i]
                      + Pb[1 * DIM * DIM + i]
                      + Pb[2 * DIM * DIM + i]
                      + Pb[3 * DIM * DIM + i];
        const int e = i >> 6;
        const int d = i & 63;
        Ms[e * LDSW + d] = s;
    }
    __syncthreads();

    const int jt  = jtb + wave;
    const float* __restrict__ X = x + (size_t)bh * TLEN * DIM;
    const int row = jt * 16 + l;            // A: M index = output row j

    v8f acc[4] = {};                        // 4 d-tiles of the 16x64 output strip

    #pragma unroll
    for (int e = 0; e < DIM; e += 4) {
        const int e0 = e + 2 * half;        // this lane-half's K pair
        // A: aligned float2 of X[row, e0..e0+1]
        const v2f a = *(const v2f*)(X + row * DIM + e0);
        #pragma unroll
        for (int dt = 0; dt < 4; ++dt) {
            v2f b;
            b.x = Ms[e0 * LDSW + dt * 16 + l];
            b.y = Ms[(e0 + 1) * LDSW + dt * 16 + l];
            acc[dt] = __builtin_amdgcn_wmma_f32_16x16x4_f32(false, a, false, b,
                                                            (short)0, acc[dt], false, false);
        }
    }

    float* __restrict__ O = out + (size_t)bh * TLEN * DIM;
    #pragma unroll
    for (int dt = 0; dt < 4; ++dt) {
        #pragma unroll
        for (int r = 0; r < 8; ++r) {
            const int j = jt * 16 + r + 8 * half;
            O[j * DIM + dt * 16 + l] = acc[dt][r];
        }
    }
}

// ---------------------------------------------------------------------------
extern "C" void kernel_launch(void* const* d_in, const int* in_sizes, int n_in,
                              void* d_out, int out_size, void* d_ws, size_t ws_size,
                              hipStream_t stream)
{
    (void)in_sizes; (void)n_in; (void)out_size; (void)ws_size;
    const float* x = (const float*)d_in[0];
    float* outp    = (float*)d_out;
    float* P       = (float*)d_ws;   // 32 * 4 * 64 * 64 * 4B = 2 MB of scratch

    // Phase 1: partial X^T X per (bh, j-slice). 128 blocks x 16 waves.
    xtx_partial_kernel<<<BH * SLICES, 512, 0, stream>>>(x, P);
    // Phase 2: out = X @ (sum of partials). 512 blocks x 8 waves.
    xm_kernel<<<BH * 16, 256, 0, stream>>>(x, P, outp);
}